// GraphPredictor_75041668596276
// MI455X (gfx1250) — compile-verified
//
#include <hip/hip_runtime.h>
#include <hip/hip_bf16.h>

// ---------------------------------------------------------------------------
// GraphPredictor for MI455X (gfx1250, wave32)
//   xw = x @ W            -> WMMA f32 16x16x4 (full fp32 precision)
//   deg / dinv            -> atomic scatter + rsqrt
//   agg[dst] += norm*xw[src] -> wave-per-edge, float4 gather, f32 atomics (L2)
//   h = relu(agg + dinv^2*xw + b); mean-pool by batch -> atomics
//   logits = [gmean | edge_pool] @ Wp + bp
// ---------------------------------------------------------------------------

typedef __attribute__((ext_vector_type(2))) float v2f;
typedef __attribute__((ext_vector_type(8))) float v8f;

#define D_HID 128

// ---- zero the accumulation regions of the workspace (graph-replay safe) ----
__global__ void zero_ws_kernel(float* __restrict__ p, size_t nfloats) {
    size_t i = (size_t)blockIdx.x * blockDim.x + threadIdx.x;
    size_t stride = (size_t)gridDim.x * blockDim.x;
    for (; i < nfloats; i += stride) p[i] = 0.0f;
}

// ---- deg[dst] += w ---------------------------------------------------------
__global__ void deg_kernel(const int* __restrict__ dst, const float* __restrict__ w,
                           float* __restrict__ deg, int nE) {
    int e = blockIdx.x * blockDim.x + threadIdx.x;
    if (e < nE) atomicAdd(&deg[dst[e]], w[e]);
}

// ---- deg -> dinv = rsqrt(deg + 1) (in place) -------------------------------
__global__ void dinv_kernel(float* __restrict__ deg, int n) {
    int i = blockIdx.x * blockDim.x + threadIdx.x;
    if (i < n) deg[i] = rsqrtf(deg[i] + 1.0f);
}

// ---- xw = x @ W via V_WMMA_F32_16X16X4_F32 ---------------------------------
// block = 256 threads (8 wave32). Block computes rows [16*bid, 16*bid+16),
// all 128 cols. 16x128 slab of x staged in LDS, shared by the 8 waves.
// Wave w computes the 16x16 tile at cols [16w, 16w+16).
// Epilogue: C tiles staged back through LDS so global stores are coalesced
// b128 writes instead of per-row guarded scalar stores.
__global__ void gemm_xw_kernel(const float* __restrict__ x, const float* __restrict__ W,
                               float* __restrict__ xw, int n) {
    __shared__ float As[16 * D_HID];   // 8 KB, reused for A staging then C staging

    const int tid  = threadIdx.x;
    const int row0 = blockIdx.x * 16;

    const int base = tid * 8;          // 256*8 = 2048 = 16*128
    const int brow = base >> 7;        // row within 16-row slab
    const int bcol = base & 127;

    // cooperative load of 16x128 slab of x into LDS (8 floats per thread)
    {
        const float* src = x + (size_t)(row0 + brow) * D_HID + bcol;
        bool ok = (row0 + brow) < n;
        #pragma unroll
        for (int q = 0; q < 8; ++q)
            As[base + q] = ok ? src[q] : 0.0f;
    }
    __syncthreads();

    const int wave = tid >> 5;
    const int lane = tid & 31;
    const int lrow = lane & 15;    // M index (or N index for B/C)
    const int hi   = lane >> 4;    // half-wave selector
    const int col0 = wave * 16;

    v8f acc = {};
    #pragma unroll 4
    for (int k = 0; k < D_HID; k += 4) {
        const int ka = k + hi * 2;     // A: hi half-wave holds K+2,K+3 (ISA 7.12.2)
        v2f a, b;
        a.x = As[lrow * D_HID + ka];
        a.y = As[lrow * D_HID + ka + 1];
        b.x = W[(size_t)ka       * D_HID + col0 + lrow];
        b.y = W[(size_t)(ka + 1) * D_HID + col0 + lrow];
        // 8 args: (neg_a, A, neg_b, B, c_mod, C, reuse_a, reuse_b)
        acc = __builtin_amdgcn_wmma_f32_16x16x4_f32(
                false, a, false, b, (short)0, acc, false, false);
    }

    // stage C tile back into LDS in natural row-major layout
    // C/D layout: VGPR r -> (M=r, N=lane) for lanes 0-15, (M=r+8, N=lane-16)
    __syncthreads();                    // all waves done reading As
    {
        const int ccol = col0 + lrow;
        #pragma unroll
        for (int r = 0; r < 8; ++r)
            As[(r + hi * 8) * D_HID + ccol] = acc[r];
    }
    __syncthreads();

    // coalesced store: each thread writes 8 contiguous floats (2 x b128)
    if (row0 + brow < n) {
        float4* dstp = (float4*)(xw + (size_t)(row0 + brow) * D_HID + bcol);
        const float4* srcp = (const float4*)(As + base);
        dstp[0] = srcp[0];
        dstp[1] = srcp[1];
    }
}

// ---- agg[dst] += (dinv[src]*w*dinv[dst]) * xw[src] -------------------------
// one wave32 per edge; lane handles channels [4*lane, 4*lane+4)
__global__ void edge_agg_kernel(const int* __restrict__ src, const int* __restrict__ dst,
                                const float* __restrict__ w, const float* __restrict__ dinv,
                                const float* __restrict__ xw, float* __restrict__ agg,
                                int nE) {
    const int lane = threadIdx.x & 31;
    const int e = (int)(((size_t)blockIdx.x * blockDim.x + threadIdx.x) >> 5);
    if (e >= nE) return;

    const int s = src[e];
    const int d = dst[e];
    const float norm = dinv[s] * w[e] * dinv[d];

    const float4 v = ((const float4*)(xw + (size_t)s * D_HID))[lane];
    float* ad = agg + (size_t)d * D_HID + lane * 4;
    atomicAdd(ad + 0, norm * v.x);
    atomicAdd(ad + 1, norm * v.y);
    atomicAdd(ad + 2, norm * v.z);
    atomicAdd(ad + 3, norm * v.w);
}

// ---- h = relu(agg + dinv^2*xw + b); pool sums/cnt by batch -----------------
// one block (128 threads) per node; thread = channel
__global__ void finalize_pool_kernel(const float* __restrict__ xw, const float* __restrict__ dinv,
                                     const float* __restrict__ b, const float* __restrict__ agg,
                                     const int* __restrict__ batch,
                                     float* __restrict__ sums, float* __restrict__ cnt, int n) {
    const int i = blockIdx.x;
    if (i >= n) return;
    const int c = threadIdx.x;
    const float di = dinv[i];
    float h = agg[(size_t)i * D_HID + c] + di * di * xw[(size_t)i * D_HID + c] + b[c];
    h = fmaxf(h, 0.0f);
    const int g = batch[i];
    atomicAdd(&sums[g * D_HID + c], h);
    if (c == 0) atomicAdd(&cnt[g], 1.0f);
}

// ---- logits = [sums/cnt | edge_pool] @ Wp + bp -----------------------------
// one block per graph; threads 0,1 compute the two classes
__global__ void head_kernel(const float* __restrict__ sums, const float* __restrict__ cnt,
                            const float* __restrict__ edge_pool, const float* __restrict__ Wp,
                            const float* __restrict__ bp, float* __restrict__ out, int nCls) {
    const int g = blockIdx.x;
    const int j = threadIdx.x;
    if (j >= nCls) return;
    const float inv = 1.0f / fmaxf(cnt[g], 1.0f);
    float acc = bp[j];
    for (int k = 0; k < D_HID; ++k)
        acc = fmaf(sums[g * D_HID + k] * inv, Wp[k * nCls + j], acc);
    for (int k = 0; k < D_HID; ++k)
        acc = fmaf(edge_pool[g * D_HID + k], Wp[(D_HID + k) * nCls + j], acc);
    out[g * nCls + j] = acc;
}

// ---------------------------------------------------------------------------
extern "C" void kernel_launch(void* const* d_in, const int* in_sizes, int n_in,
                              void* d_out, int out_size, void* d_ws, size_t ws_size,
                              hipStream_t stream) {
    const float* x         = (const float*)d_in[0];
    const int*   edge_idx  = (const int*)  d_in[1];   // [2, E] flat
    const float* weights   = (const float*)d_in[2];
    const int*   batch     = (const int*)  d_in[3];
    const float* edge_pool = (const float*)d_in[4];
    const float* W         = (const float*)d_in[5];
    const float* b         = (const float*)d_in[6];
    const float* Wp        = (const float*)d_in[7];
    const float* bp        = (const float*)d_in[8];

    const int n  = in_sizes[0] / D_HID;       // 100000 nodes
    const int nE = in_sizes[2];               // 2000000 edges
    const int G  = in_sizes[4] / D_HID;       // 128 graphs
    const int nCls = out_size / G;            // 2 classes

    const int* src = edge_idx;
    const int* dst = edge_idx + nE;

    // workspace layout: [deg | agg | sums | cnt] (zeroed) then [xw]
    float* ws   = (float*)d_ws;
    float* deg  = ws;                                   // n
    float* agg  = deg  + n;                             // n*128
    float* sums = agg  + (size_t)n * D_HID;             // G*128
    float* cnt  = sums + (size_t)G * D_HID;             // G
    float* xw   = cnt  + G;                             // n*128
    const size_t nzero = (size_t)n + (size_t)n * D_HID + (size_t)G * D_HID + G;

    zero_ws_kernel<<<2048, 256, 0, stream>>>(ws, nzero);

    deg_kernel<<<(nE + 255) / 256, 256, 0, stream>>>(dst, weights, deg, nE);
    dinv_kernel<<<(n + 255) / 256, 256, 0, stream>>>(deg, n);

    gemm_xw_kernel<<<(n + 15) / 16, 256, 0, stream>>>(x, W, xw, n);

    // one wave32 per edge -> nE*32 threads
    {
        size_t threads = (size_t)nE * 32;
        int blocks = (int)((threads + 255) / 256);
        edge_agg_kernel<<<blocks, 256, 0, stream>>>(src, dst, weights, deg, xw, agg, nE);
    }

    finalize_pool_kernel<<<n, D_HID, 0, stream>>>(xw, deg, b, agg, batch, sums, cnt, n);

    head_kernel<<<G, 32, 0, stream>>>(sums, cnt, edge_pool, Wp, bp, (float*)d_out, nCls);
}